// SimpleHANFallback_70136815944150
// MI455X (gfx1250) — compile-verified
//
#include <hip/hip_runtime.h>

// ---- problem constants (from reference) ----
#define N_Pn 50000
#define N_Wn 10000
#define D_PF 256
#define D_WF 128
#define HIDn 64
#define OUTn 8
#define KHn  4
#define LAYn 2
#define E_PPn 500000
#define E_PWn 150000
#define E_WPn 150000

#define MBT 8            // m-tiles (16 rows each) per block
#define BSH_ELEMS 16384  // 64 KB LDS B panel: Kd*N <= 16384 for every GEMM here

typedef float v2f __attribute__((ext_vector_type(2)));
typedef float v8f __attribute__((ext_vector_type(8)));
typedef unsigned int u32x4 __attribute__((ext_vector_type(4)));
typedef int i32x4 __attribute__((ext_vector_type(4)));
typedef int i32x8 __attribute__((ext_vector_type(8)));

// ------------------------------------------------------------------
// Tensor Data Mover: DMA the whole B panel [Kd x N] f32 (row-major,
// contiguous) from global memory into LDS at byte offset ldsOff.
// D# built per CDNA5 ISA ch.8 (groups 0/1; groups 2/3 zero => 2D tensor).
// This toolchain exposes the 6-arg builtin:
//   (u32x4 g0, i32x8 g1, i32x4 g2, i32x4 g3, i32x8 ext, i32 cpol)
// ------------------------------------------------------------------
__device__ __forceinline__ void tdm_load_B(const float* gB, unsigned ldsOff,
                                           int Kd, int N) {
  unsigned long long ga = (unsigned long long)gB;
  u32x4 g0;
  g0[0] = 1u;                                    // count=1, user mode, no gather
  g0[1] = ldsOff;                                // lds_addr (bytes)
  g0[2] = (unsigned)(ga & 0xffffffffu);          // global_addr[31:0]
  g0[3] = (unsigned)((ga >> 32) & 0x01ffffffu)   // global_addr[56:32]
          | (2u << 30);                          // type = 2 ("image")
  i32x8 g1;
  g1[0] = (2 << 16);                             // data_size=2 -> 4-byte elems
  g1[1] = (int)((unsigned)N << 16);              // tensor_dim0[15:0]
  g1[2] = (int)((((unsigned)N >> 16) & 0xffffu)  // tensor_dim0[31:16]
          | ((unsigned)Kd << 16));               // tensor_dim1[15:0]
  g1[3] = (int)((((unsigned)Kd >> 16) & 0xffffu) // tensor_dim1[31:16]
          | ((unsigned)N << 16));                // tile_dim0 = N
  g1[4] = (int)(unsigned)Kd;                     // tile_dim1 = Kd (tile_dim2=0)
  g1[5] = (int)(unsigned)N;                      // tensor_dim0_stride[31:0]
  g1[6] = 0;                                     // stride hi bits / dim1_stride
  g1[7] = 0;
  i32x4 z4 = {0, 0, 0, 0};
  i32x8 z8 = {0, 0, 0, 0, 0, 0, 0, 0};
  __builtin_amdgcn_tensor_load_to_lds(g0, g1, z4, z4, z8, 0);
}

// ------------------------------------------------------------------
// f32 WMMA GEMM (N, Kd compile-time): C[M,N] = A[M,Kd] @ B[Kd,N] (+bias[n]).
// B panel staged once per block into LDS by the TDM; each block computes
// MBT m-tiles x all n-tiles (8 waves stride over tiles). Compile-time N/Kd
// turn all LDS offsets into immediates and fully determine the k-loop.
// Optional fused attention epilogue (GAT a_s / a_d):
//   avec[m*KHn + head] += sum_n C[m,n] * att[head*HIDn + n%64]
// Register layouts per ISA 7.12.2:
//   A 16x4 f32 : row = lane&15, K = 2*(lane>>4) + e     (e = 0,1)
//   B  4x16 f32: col = lane&15, K = 2*(lane>>4) + e
//   C 16x16 f32: M = e + 8*(lane>>4), N = lane&15       (e = 0..7)
// ------------------------------------------------------------------
template <int Nc, int Kc>
__global__ __launch_bounds__(256)
void wmma_gemm_f32(const float* __restrict__ A, const float* __restrict__ B,
                   float* __restrict__ C, const float* __restrict__ bias,
                   const float* __restrict__ att, float* __restrict__ avec,
                   int M) {
  __shared__ float Bsh[BSH_ELEMS];
  constexpr int tilesN = Nc >> 4;
  const int lane = threadIdx.x & 31;
  const int wave = threadIdx.x >> 5;
  const int tilesM = M >> 4;

  if (threadIdx.x == 0) {                    // one TDM DMA per block
    tdm_load_B(B, (unsigned)(unsigned long long)&Bsh[0], Kc, Nc);
    __builtin_amdgcn_s_wait_tensorcnt(0);
  }
  __syncthreads();

  const int r = lane & 15;
  const int g = lane >> 4;
  const int mtBase = blockIdx.x * MBT;

  for (int it = wave; it < MBT * tilesN; it += 8) {
    const int mt = mtBase + it / tilesN;
    if (mt >= tilesM) continue;              // wave-uniform guard
    const int m0 = mt << 4;
    const int n0 = (it % tilesN) << 4;

    const float* Arow = A + (size_t)(m0 + r) * (size_t)Kc;
    const float* Bcol = Bsh + n0 + r + 2 * g * Nc;   // K = 2g base
    v8f acc = {};
#pragma unroll 8
    for (int k0 = 0; k0 < Kc; k0 += 4) {
      v2f a, b;
      a.x = Arow[k0 + 2 * g + 0];
      a.y = Arow[k0 + 2 * g + 1];
      b.x = Bcol[(k0 + 0) * Nc];
      b.y = Bcol[(k0 + 1) * Nc];
      acc = __builtin_amdgcn_wmma_f32_16x16x4_f32(false, a, false, b,
                                                  (short)0, acc, false, false);
    }

    const int nn = n0 + r;
#pragma unroll
    for (int e = 0; e < 8; ++e) {
      const int mm = m0 + e + 8 * g;
      float v = acc[e];
      if (bias) v += bias[nn];
      if (C) C[(size_t)mm * (size_t)Nc + nn] = v;
      if (att) {
        const int head = nn >> 6;
        const int h    = nn & 63;
        atomicAdd(&avec[mm * KHn + head], v * att[head * HIDn + h]);
      }
    }
  }
}

// ---- monotonic float<->uint key for atomic float max ----
__device__ __forceinline__ unsigned fkey(float f) {
  unsigned u = __float_as_uint(f);
  return (u & 0x80000000u) ? ~u : (u | 0x80000000u);
}
__device__ __forceinline__ float funkey(unsigned k) {
  unsigned u = (k & 0x80000000u) ? (k & 0x7fffffffu) : ~k;
  return __uint_as_float(u);
}

// Pass A: per (edge, head) leaky-relu logit; segment max via atomicMax(key)
__global__ void edge_logits(const int* __restrict__ src, const int* __restrict__ dst,
                            const float* __restrict__ a_s, const float* __restrict__ a_d,
                            float* __restrict__ ebuf, unsigned* __restrict__ mkey, int E) {
  int i = blockIdx.x * blockDim.x + threadIdx.x;
  if (i >= E * KHn) return;
  const int e = i >> 2, k = i & 3;
  const int s = src[e], d = dst[e];
  float v = a_s[s * KHn + k] + a_d[d * KHn + k];
  v = v > 0.f ? v : 0.2f * v;
  ebuf[i] = v;
  atomicMax(&mkey[d * KHn + k], fkey(v));
}

// Pass B: exp(e - max) and segment-sum denominator
__global__ void edge_exp(const int* __restrict__ dst, float* __restrict__ ebuf,
                         const unsigned* __restrict__ mkey,
                         float* __restrict__ denom, int E) {
  int i = blockIdx.x * blockDim.x + threadIdx.x;
  if (i >= E * KHn) return;
  const int e = i >> 2, k = i & 3;
  const int d = dst[e];
  float ex = __expf(ebuf[i] - funkey(mkey[d * KHn + k]));
  ebuf[i] = ex;
  atomicAdd(&denom[d * KHn + k], ex);
}

// Pass C: weighted message scatter. One block (256 threads) per edge:
// coalesced 256-float gather of hs[src], coalesced atomics into acc[dst].
__global__ __launch_bounds__(256)
void edge_scatter(const int* __restrict__ src, const int* __restrict__ dst,
                  const float* __restrict__ ebuf, const float* __restrict__ denom,
                  const float* __restrict__ hs, float* __restrict__ acc, int E) {
  const int i = blockIdx.x * 256 + threadIdx.x;
  if ((i >> 8) >= E) return;
  const int e = i >> 8, c = i & 255, k = c >> 6;
  const int s = src[e], d = dst[e];
  const float alpha = ebuf[e * KHn + k] / denom[d * KHn + k];
  atomicAdd(&acc[(size_t)d * 256 + c], hs[(size_t)s * 256 + c] * alpha);
}

// Head-mean + bias(es) + scale + ELU
__global__ void combine(const float* __restrict__ acc, const float* __restrict__ bias_a,
                        const float* __restrict__ bias_b, float scale,
                        float* __restrict__ hout, int Nn) {
  int i = blockIdx.x * blockDim.x + threadIdx.x;
  if (i >= Nn * HIDn) return;
  const int n = i >> 6, h = i & 63;
  const float* a = acc + (size_t)n * 256;
  float v = 0.25f * (a[h] + a[64 + h] + a[128 + h] + a[192 + h]) + bias_a[h];
  if (bias_b) v += bias_b[h];
  v *= scale;
  hout[i] = v > 0.f ? v : (__expf(v) - 1.f);
}

// Tiny post GEMM [Nn,64] @ [64,8] + b  (memory bound; scalar is fine)
__global__ void post_gemm(const float* __restrict__ h, const float* __restrict__ W,
                          const float* __restrict__ b, float* __restrict__ out, int Nn) {
  int i = blockIdx.x * blockDim.x + threadIdx.x;
  if (i >= Nn * OUTn) return;
  const int n = i >> 3, o = i & 7;
  const float* hr = h + (size_t)n * HIDn;
  float v = b[o];
#pragma unroll
  for (int k = 0; k < HIDn; ++k) v += hr[k] * W[k * OUTn + o];
  out[i] = v;
}

static inline int cdiv(long long a, long long b) { return (int)((a + b - 1) / b); }

extern "C" void kernel_launch(void* const* d_in, const int* in_sizes, int n_in,
                              void* d_out, int out_size, void* d_ws, size_t ws_size,
                              hipStream_t stream) {
  const float* x_p      = (const float*)d_in[0];
  const float* x_w      = (const float*)d_in[1];
  const float* W_proj_p = (const float*)d_in[2];
  const float* b_proj_p = (const float*)d_in[3];
  const float* W_proj_w = (const float*)d_in[4];
  const float* b_proj_w = (const float*)d_in[5];
  const float* W_src    = (const float*)d_in[6];   // [L,3,64,256]
  const float* W_dst    = (const float*)d_in[7];   // [L,3,64,256]
  const float* att_src  = (const float*)d_in[8];   // [L,3,4,64]
  const float* att_dst  = (const float*)d_in[9];   // [L,3,4,64]
  const float* conv_b   = (const float*)d_in[10];  // [L,3,64]
  const float* W_post_p = (const float*)d_in[11];
  const float* b_post_p = (const float*)d_in[12];
  const float* W_post_w = (const float*)d_in[13];
  const float* b_post_w = (const float*)d_in[14];
  const int*   ei_pp    = (const int*)d_in[15];    // [2, E_PP]
  const int*   ei_pw    = (const int*)d_in[16];
  const int*   ei_wp    = (const int*)d_in[17];
  float* out = (float*)d_out;

  // ---- workspace carve-up (~155 MB of f32) ----
  float* w = (float*)d_ws;
  float* h_p0 = w;  w += (size_t)N_Pn * HIDn;
  float* h_w0 = w;  w += (size_t)N_Wn * HIDn;
  float* h_p1 = w;  w += (size_t)N_Pn * HIDn;
  float* h_w1 = w;  w += (size_t)N_Wn * HIDn;
  float* hs   = w;  w += (size_t)N_Pn * 256;
  float* a_s  = w;  w += (size_t)N_Pn * KHn;
  float* a_d  = w;  w += (size_t)N_Pn * KHn;
  unsigned* mkey = (unsigned*)w; w += (size_t)N_Pn * KHn;
  float* denom = w; w += (size_t)N_Pn * KHn;
  float* ebuf  = w; w += (size_t)E_PPn * KHn;
  float* acc_p = w; w += (size_t)N_Pn * 256;
  float* acc_w = w; w += (size_t)N_Wn * 256;
  (void)ws_size; (void)in_sizes; (void)n_in; (void)out_size;

  // ---- input projections (WMMA + TDM-staged B) ----
  wmma_gemm_f32<HIDn, D_PF><<<cdiv(N_Pn / 16, MBT), 256, 0, stream>>>(
      x_p, W_proj_p, h_p0, b_proj_p, nullptr, nullptr, N_Pn);
  wmma_gemm_f32<HIDn, D_WF><<<cdiv(N_Wn / 16, MBT), 256, 0, stream>>>(
      x_w, W_proj_w, h_w0, b_proj_w, nullptr, nullptr, N_Wn);

  float* hp = h_p0;  float* hw = h_w0;
  float* hpn = h_p1; float* hwn = h_w1;

  for (int l = 0; l < LAYn; ++l) {
    (void)hipMemsetAsync(acc_p, 0, (size_t)N_Pn * 256 * sizeof(float), stream);
    (void)hipMemsetAsync(acc_w, 0, (size_t)N_Wn * 256 * sizeof(float), stream);

    for (int t = 0; t < 3; ++t) {
      const float* hsrc = (t == 2) ? hw : hp;
      const float* hdst = (t == 1) ? hw : hp;
      const int Ns = (t == 2) ? N_Wn : N_Pn;
      const int Nd = (t == 1) ? N_Wn : N_Pn;
      const int* ep = (t == 0) ? ei_pp : (t == 1) ? ei_pw : ei_wp;
      const int E  = (t == 0) ? E_PPn : (t == 1) ? E_PWn : E_WPn;
      float* acc = (t == 1) ? acc_w : acc_p;
      const size_t wo = (size_t)(l * 3 + t) * HIDn * (KHn * HIDn);
      const size_t ao = (size_t)(l * 3 + t) * KHn * HIDn;
      const int* esrc = ep;
      const int* edst = ep + E;

      (void)hipMemsetAsync(a_s,   0, (size_t)Ns * KHn * sizeof(float), stream);
      (void)hipMemsetAsync(a_d,   0, (size_t)Nd * KHn * sizeof(float), stream);
      (void)hipMemsetAsync(mkey,  0, (size_t)Nd * KHn * sizeof(unsigned), stream);
      (void)hipMemsetAsync(denom, 0, (size_t)Nd * KHn * sizeof(float), stream);

      // hs = hsrc @ W_src[l,t]   (stored) ; fused a_s accumulation
      wmma_gemm_f32<KHn * HIDn, HIDn><<<cdiv(Ns / 16, MBT), 256, 0, stream>>>(
          hsrc, W_src + wo, hs, nullptr, att_src + ao, a_s, Ns);
      // hd = hdst @ W_dst[l,t]   (NOT stored) ; only fused a_d accumulation
      wmma_gemm_f32<KHn * HIDn, HIDn><<<cdiv(Nd / 16, MBT), 256, 0, stream>>>(
          hdst, W_dst + wo, nullptr, nullptr, att_dst + ao, a_d, Nd);

      edge_logits<<<cdiv((long long)E * KHn, 256), 256, 0, stream>>>(
          esrc, edst, a_s, a_d, ebuf, mkey, E);
      edge_exp<<<cdiv((long long)E * KHn, 256), 256, 0, stream>>>(
          edst, ebuf, mkey, denom, E);
      edge_scatter<<<E, 256, 0, stream>>>(esrc, edst, ebuf, denom, hs, acc, E);
    }

    // HeteroConv mean + ELU
    combine<<<cdiv((long long)N_Pn * HIDn, 256), 256, 0, stream>>>(
        acc_p, conv_b + (size_t)(l * 3 + 0) * HIDn,
               conv_b + (size_t)(l * 3 + 2) * HIDn, 0.5f, hpn, N_Pn);
    combine<<<cdiv((long long)N_Wn * HIDn, 256), 256, 0, stream>>>(
        acc_w, conv_b + (size_t)(l * 3 + 1) * HIDn, nullptr, 1.0f, hwn, N_Wn);

    float* tmp;
    tmp = hp; hp = hpn; hpn = tmp;
    tmp = hw; hw = hwn; hwn = tmp;
  }

  post_gemm<<<cdiv((long long)N_Pn * OUTn, 256), 256, 0, stream>>>(
      hp, W_post_p, b_post_p, out, N_Pn);
  post_gemm<<<cdiv((long long)N_Wn * OUTn, 256), 256, 0, stream>>>(
      hw, W_post_w, b_post_w, out + (size_t)N_Pn * OUTn, N_Wn);
}